// JAXFastRnn_30202210025728
// MI455X (gfx1250) — compile-verified
//
#include <hip/hip_runtime.h>
#include <math.h>

typedef float v2f __attribute__((ext_vector_type(2)));
typedef float v8f __attribute__((ext_vector_type(8)));

#define Cc 8
#define Rr 7
#define Jd 2048
#define Kd 2048
#define JKsz ((size_t)Jd * (size_t)Kd)
#define JCHUNKS 32
#define JB (Jd / JCHUNKS)   // 64 rows of j per block
#define PFDIST 16           // prefetch distance (j rows)

// ---------------------------------------------------------------------------
// single-instruction tanh (v_tanh_f32 on gfx1250), branch-free fallback
// ---------------------------------------------------------------------------
__device__ __forceinline__ float fast_tanh(float x) {
#if __has_builtin(__builtin_amdgcn_tanhf)
    return __builtin_amdgcn_tanhf(x);
#elif __has_builtin(__builtin_amdgcn_tanh_f32)
    return __builtin_amdgcn_tanh_f32(x);
#else
    float e = __expf(2.0f * x);
    return 1.0f - 2.0f * __builtin_amdgcn_rcpf(e + 1.0f);
#endif
}

// ---------------------------------------------------------------------------
// init: zero h[J] and norms2[K] in workspace
// ---------------------------------------------------------------------------
__global__ void k_init(float* __restrict__ h, float* __restrict__ n2) {
    int i = blockIdx.x * blockDim.x + threadIdx.x;
    if (i < Jd) h[i] = 0.0f;
    if (i < Kd) n2[i] = 0.0f;
}

// ---------------------------------------------------------------------------
// GEMV: h[j] = sum_k post_act[k] * weight[k, j]   (weight is (K, J) row-major)
// ---------------------------------------------------------------------------
__global__ void k_gemv(const float* __restrict__ weight,
                       const float* __restrict__ post_act,
                       float* __restrict__ h) {
    int j  = blockIdx.x * 256 + threadIdx.x;
    int k0 = blockIdx.y * 128;
    float acc = 0.0f;
#pragma unroll 8
    for (int kk = 0; kk < 128; ++kk) {
        int k = k0 + kk;
        acc = fmaf(post_act[k], weight[(size_t)k * Jd + j], acc);
    }
    atomicAdd(&h[j], acc);
}

// ---------------------------------------------------------------------------
// Main body. Each wave owns 16 consecutive k positions, loops over JB j's.
// drive = G(16x16) x B(16x16) via 4 chained V_WMMA_F32_16X16X4_F32.
// inner rows: 0..6 = rules u0,u2,u4,u9,u10,u11,u12 ; 7..14 = sw[0..7] ; 15 = 0
// G row permutation: rows 0..3 = chems {1,2,5,6} (low-half lanes' chems),
//                    rows 8..11 = chems {0,3,4,7} (high-half lanes' chems),
//                    other rows 0.  =>  every lane's acc[0..3] is exactly the
// drive for the 4 chemicals whose sw values it loaded — no cross-lane swaps.
// Cache policy: norms pass reads sw with RT hints (seed 192MB L2 for reuse);
// emit pass uses NT loads (last use) and NT stores (write-once outputs) so
// the streamed-out 144MB does not evict the L2-resident sw mid-pass.
// ---------------------------------------------------------------------------
template <bool EMIT>
__device__ __forceinline__ void main_body(
    const float* __restrict__ sw,   const float* __restrict__ pre_act,
    const float* __restrict__ post_act, const float* __restrict__ pre_err,
    const float* __restrict__ post_err, const float* __restrict__ Q,
    const float* __restrict__ Km,   const float* __restrict__ zv,
    const float* __restrict__ yv,   const float* __restrict__ vv,
    const float* __restrict__ h,    float* __restrict__ norms2,
    float* __restrict__ out_pw,     float* __restrict__ out_sw) {

    const int tid  = threadIdx.x;
    const int lane = tid & 31;
    const int wave = tid >> 5;
    const bool lo  = (lane < 16);
    const int kpos = (blockIdx.x * 8 + wave) * 16 + (lane & 15);
    const int j0   = blockIdx.y * JB;

    // ---- A matrix (coefficients, permuted rows), constant over the loop ----
    // A 16x4 f32 layout: lane L holds row M=L&15; lanes<16 carry cols 4q,4q+1,
    // lanes>=16 carry cols 4q+2,4q+3.
    const int m    = lane & 15;
    const int kofs = lo ? 0 : 2;
    v2f a0, a1, a2, a3;
    {
        const int lochems[4] = {1, 2, 5, 6};
        const int hichems[4] = {0, 3, 4, 7};
        int ce = -1;
        if (m < 4)                 ce = lochems[m];
        else if (m >= 8 && m < 12) ce = hichems[m - 8];
        float g[16];
#pragma unroll
        for (int col = 0; col < 16; ++col) {
            float c = 0.0f;
            if (ce >= 0) {
                if (col < Rr)      c = Q[ce * Rr + col];          // rule mix
                else if (col < 15) c = Km[(col - Rr) * Cc + ce];  // K[i][ce]
            }
            g[col] = c;
        }
        a0.x = g[0 + kofs];  a0.y = g[1 + kofs];
        a1.x = g[4 + kofs];  a1.y = g[5 + kofs];
        a2.x = g[8 + kofs];  a2.y = g[9 + kofs];
        a3.x = g[12 + kofs]; a3.y = g[13 + kofs];
    }

    // this lane's 4 chemicals (same set used for B loads and the epilogue)
    const int cA = lo ? 1 : 0, cB = lo ? 2 : 3, cC = lo ? 5 : 4, cD = lo ? 6 : 7;
    const float yA = yv[cA], yB = yv[cB], yC = yv[cC], yD = yv[cD];
    const float zA = zv[cA], zB = zv[cB], zC = zv[cC], zD = zv[cD];

    // softmax(v) weights for our 4 chemicals
    float wA, wB, wC, wD;
    {
        float vmax = vv[0];
#pragma unroll
        for (int c = 1; c < Cc; ++c) vmax = fmaxf(vmax, vv[c]);
        float e[Cc], s = 0.0f;
#pragma unroll
        for (int c = 0; c < Cc; ++c) { e[c] = __expf(vv[c] - vmax); s += e[c]; }
        const float inv = 1.0f / s;
        wA = e[cA] * inv; wB = e[cB] * inv; wC = e[cC] * inv; wD = e[cD] * inv;
    }

    const float poa = post_act[kpos];
    const float poe = post_err[kpos];
    const float rn  = EMIT ? rsqrtf(norms2[kpos]) : 1.0f;

    const size_t bA = (size_t)cA * JKsz + kpos;
    const size_t bB = (size_t)cB * JKsz + kpos;
    const size_t bC = (size_t)cC * JKsz + kpos;
    const size_t bD = (size_t)cD * JKsz + kpos;

    float sumsq = 0.0f;

#pragma unroll 2
    for (int j = j0; j < j0 + JB; ++j) {
        const float pa = pre_act[j];
        const float pe = pre_err[j];
        const float hj = h[j];
        const size_t rowoff = (size_t)j * Kd;

        // emit pass: last use of sw -> non-temporal loads
        const float sA = EMIT ? __builtin_nontemporal_load(&sw[bA + rowoff]) : sw[bA + rowoff];
        const float sB = EMIT ? __builtin_nontemporal_load(&sw[bB + rowoff]) : sw[bB + rowoff];
        const float sC = EMIT ? __builtin_nontemporal_load(&sw[bC + rowoff]) : sw[bC + rowoff];
        const float sD = EMIT ? __builtin_nontemporal_load(&sw[bD + rowoff]) : sw[bD + rowoff];

        // prefetch sw streams PFDIST rows ahead (clamped; WGP scope, all levels)
        {
            int jpf = j + PFDIST; if (jpf >= Jd) jpf = Jd - 1;
            const size_t pfo = (size_t)jpf * Kd;
            __builtin_prefetch(&sw[bA + pfo], 0, 3);
            __builtin_prefetch(&sw[bB + pfo], 0, 3);
            __builtin_prefetch(&sw[bC + pfo], 0, 3);
            __builtin_prefetch(&sw[bD + pfo], 0, 3);
        }

        // structured rule values at (j, kpos)
        const float u0v  = pa * poe;
        const float u2v  = pe * poe;
        const float u4v  = pe;
        const float u9v  = poa * (pa - hj);
        const float u10v = poe;
        const float u11v = poa;
        const float u12v = pa;

        // B (4x16) per chunk q: lanes<16 carry rows 4q,4q+1; lanes>=16 rows 4q+2,4q+3
        v2f b0, b1, b2, b3;
        b0.x = lo ? u0v  : u4v;   b0.y = lo ? u2v  : u9v;
        b1.x = lo ? u10v : u12v;  b1.y = lo ? u11v : sA;    // hi row7 = sw0
        b2.x = lo ? sA   : sB;    b2.y = lo ? sB   : sC;    // rows 8,9 | 10,11
        b3.x = lo ? sC   : sD;    b3.y = lo ? sD   : 0.0f;  // rows 12,13 | 14,0

        v8f acc = {};
        acc = __builtin_amdgcn_wmma_f32_16x16x4_f32(false, a0, false, b0, (short)0, acc, false, false);
        acc = __builtin_amdgcn_wmma_f32_16x16x4_f32(false, a1, false, b1, (short)0, acc, false, false);
        acc = __builtin_amdgcn_wmma_f32_16x16x4_f32(false, a2, false, b2, (short)0, acc, false, false);
        acc = __builtin_amdgcn_wmma_f32_16x16x4_f32(false, a3, false, b3, (short)0, acc, false, false);

        // acc[0..3] = drive for this lane's chems (cA,cB,cC,cD) — by G row perm
        const float nsA = fmaf(yA, sA, zA * fast_tanh(acc[0]));
        const float nsB = fmaf(yB, sB, zB * fast_tanh(acc[1]));
        const float nsC = fmaf(yC, sC, zC * fast_tanh(acc[2]));
        const float nsD = fmaf(yD, sD, zD * fast_tanh(acc[3]));

        float pwp = fmaf(wA, nsA, fmaf(wB, nsB, fmaf(wC, nsC, wD * nsD)));
        const float pw = pwp + __shfl_xor(pwp, 16, 32);   // both halves hold full sum

        if (EMIT) {
            // write-once outputs -> non-temporal stores (don't evict sw from L2)
            __builtin_nontemporal_store(nsA * rn, &out_sw[bA + rowoff]);
            __builtin_nontemporal_store(nsB * rn, &out_sw[bB + rowoff]);
            __builtin_nontemporal_store(nsC * rn, &out_sw[bC + rowoff]);
            __builtin_nontemporal_store(nsD * rn, &out_sw[bD + rowoff]);
            // lanes L and L+16 write the same value to the same address:
            // no EXEC-mask branch needed
            __builtin_nontemporal_store(pw * rn, &out_pw[(size_t)kpos * Jd + j]);
        } else {
            sumsq = fmaf(pw, pw, sumsq);
        }
    }

    if (!EMIT && lo) atomicAdd(&norms2[kpos], sumsq);
}

__global__ void k_norms(const float* __restrict__ sw, const float* __restrict__ pre_act,
                        const float* __restrict__ post_act, const float* __restrict__ pre_err,
                        const float* __restrict__ post_err, const float* __restrict__ Q,
                        const float* __restrict__ Km, const float* __restrict__ zv,
                        const float* __restrict__ yv, const float* __restrict__ vv,
                        const float* __restrict__ h, float* __restrict__ norms2) {
    main_body<false>(sw, pre_act, post_act, pre_err, post_err, Q, Km, zv, yv, vv,
                     h, norms2, nullptr, nullptr);
}

__global__ void k_emit(const float* __restrict__ sw, const float* __restrict__ pre_act,
                       const float* __restrict__ post_act, const float* __restrict__ pre_err,
                       const float* __restrict__ post_err, const float* __restrict__ Q,
                       const float* __restrict__ Km, const float* __restrict__ zv,
                       const float* __restrict__ yv, const float* __restrict__ vv,
                       const float* __restrict__ h, float* __restrict__ norms2,
                       float* __restrict__ out_pw, float* __restrict__ out_sw) {
    main_body<true>(sw, pre_act, post_act, pre_err, post_err, Q, Km, zv, yv, vv,
                    h, norms2, out_pw, out_sw);
}

// ---------------------------------------------------------------------------
extern "C" void kernel_launch(void* const* d_in, const int* in_sizes, int n_in,
                              void* d_out, int out_size, void* d_ws, size_t ws_size,
                              hipStream_t stream) {
    const float* sw       = (const float*)d_in[0];   // (C, J, K)
    const float* weight   = (const float*)d_in[1];   // (K, J)
    const float* pre_act  = (const float*)d_in[2];   // (J,)
    const float* post_act = (const float*)d_in[3];   // (K,)
    const float* pre_err  = (const float*)d_in[4];   // (J,)
    const float* post_err = (const float*)d_in[5];   // (K,)
    const float* Q        = (const float*)d_in[6];   // (C, R)
    const float* Km       = (const float*)d_in[7];   // (C, C)
    const float* zv       = (const float*)d_in[8];   // (C,)
    const float* yv       = (const float*)d_in[9];   // (C,)
    const float* vv       = (const float*)d_in[10];  // (C,)

    float* out_pw = (float*)d_out;                   // (K, J)
    float* out_sw = out_pw + JKsz;                   // (C, J, K)

    float* h      = (float*)d_ws;                    // J floats
    float* norms2 = h + Jd;                          // K floats

    k_init<<<dim3((Jd + 255) / 256), dim3(256), 0, stream>>>(h, norms2);
    k_gemv<<<dim3(Jd / 256, 16), dim3(256), 0, stream>>>(weight, post_act, h);

    dim3 grid(Kd / (8 * 16), JCHUNKS);   // (16, 32)
    dim3 block(256);                     // 8 waves, one 16-wide k tile each
    k_norms<<<grid, block, 0, stream>>>(sw, pre_act, post_act, pre_err, post_err,
                                        Q, Km, zv, yv, vv, h, norms2);
    k_emit<<<grid, block, 0, stream>>>(sw, pre_act, post_act, pre_err, post_err,
                                       Q, Km, zv, yv, vv, h, norms2, out_pw, out_sw);
}